// CachedAttention_15882789060757
// MI455X (gfx1250) — compile-verified
//
#include <hip/hip_runtime.h>
#include <math.h>

// ---------------------------------------------------------------------------
// CachedAttention decode step, MI455X (gfx1250, wave32).
//   B=16, T=1, D=4096, H=32, DK=128, S=2048.
// Memory-bound: ~1.33 GB mandatory traffic -> ~57us at 23.3 TB/s.
// Stage 1: q/k_new/v_new = x @ W{q,k,v}^T   fused, WMMA f32 16x16x4,
//          K-split x8 per block (6144 waves -> enough loads in flight).
// Stage 2: flash-decode attention per (b,h), coalesced b128 streaming,
//          deterministic LDS tree combines (no float atomics).
// Stage 3: out = attn_out @ Wo^T            same WMMA GEMM.
// ---------------------------------------------------------------------------

typedef __attribute__((ext_vector_type(2))) float v2f;
typedef __attribute__((ext_vector_type(8))) float v8f;

#define D_MODEL 4096
#define NB      16
#define NH      32
#define DKH     128
#define SMAX    2048
#define SCALE   0.08838834764831845f   // 1/sqrt(128)

// ---------------------------------------------------------------------------
// OUT[16][N] = X[16][K] @ W[N][K]^T.
// One BLOCK computes one 16x16 output tile: wave w accumulates K-chunk
// [w*K/8, (w+1)*K/8) with V_WMMA_F32_16X16X4_F32, then the 8 partial tiles
// are summed in fixed order through LDS (deterministic).
// 32-bit A (16x4): lanes 0-15 -> M=0..15 carrying (K,K+1); lanes 16-31 carry
// (K+2,K+3). B (4x16) mirrored on N. C/D: VGPR r -> M=r / M=r+8, N=lane%16.
// ---------------------------------------------------------------------------
__device__ __forceinline__
void gemm16_tile(const float* __restrict__ X, const float* __restrict__ W,
                 float* __restrict__ OUT, int N, int K) {
  __shared__ float part[8 * 256];      // 8 wave-partials of the 16x16 tile

  const int tid  = threadIdx.x;
  const int lane = tid & 31;
  const int wave = tid >> 5;
  const int n0   = blockIdx.x * 16;

  const int half = lane >> 4;          // 0 -> K+0/K+1, 1 -> K+2/K+3
  const int l    = lane & 15;          // A row (M) / B column (N)
  const int kc   = K >> 3;             // per-wave K chunk (512)
  const int k0   = wave * kc;

  const float* xrow = X + (size_t)l * K + k0 + 2 * half;
  const float* wrow = W + (size_t)(n0 + l) * K + k0 + 2 * half;

  v8f acc = {0.f, 0.f, 0.f, 0.f, 0.f, 0.f, 0.f, 0.f};
  #pragma unroll 8                      // ~16 b64 loads in flight per wave
  for (int k = 0; k < kc; k += 4) {
    v2f a = *(const v2f*)(xrow + k);
    v2f b = *(const v2f*)(wrow + k);
    acc = __builtin_amdgcn_wmma_f32_16x16x4_f32(
        false, a, false, b, (short)0, acc, false, false);
  }

  #pragma unroll
  for (int r = 0; r < 8; ++r) {
    const int m = r + half * 8;
    part[wave * 256 + m * 16 + l] = acc[r];
  }
  __syncthreads();

  // fixed-order combine: thread t owns element (m = t/16, n = t%16)
  float sum = 0.f;
  #pragma unroll
  for (int w = 0; w < 8; ++w) sum += part[w * 256 + tid];
  OUT[(size_t)(tid >> 4) * N + n0 + (tid & 15)] = sum;
}

__global__ __launch_bounds__(256)
void qkv_gemm(const float* __restrict__ X,
              const float* __restrict__ Wq, const float* __restrict__ Wk,
              const float* __restrict__ Wv,
              float* __restrict__ q, float* __restrict__ kn,
              float* __restrict__ vn, int N, int K) {
  const float* W = (blockIdx.y == 0) ? Wq : (blockIdx.y == 1) ? Wk : Wv;
  float*       O = (blockIdx.y == 0) ? q  : (blockIdx.y == 1) ? kn : vn;
  gemm16_tile(X, W, O, N, K);
}

__global__ __launch_bounds__(256)
void out_gemm(const float* __restrict__ X, const float* __restrict__ W,
              float* __restrict__ O, int N, int K) {
  gemm16_tile(X, W, O, N, K);
}

// ---------------------------------------------------------------------------
// Flash-decode attention for one (b,h) per block; 256 threads = 8 waves.
// Wave w owns key/value rows s in [w*256, (w+1)*256).
// ---------------------------------------------------------------------------
__global__ __launch_bounds__(256)
void attn_decode(const float* __restrict__ q, const float* __restrict__ kc,
                 const float* __restrict__ vc, const float* __restrict__ kn,
                 const float* __restrict__ vn, const int* __restrict__ sp_p,
                 float* __restrict__ out) {
  __shared__ __align__(16) float sc[SMAX];        // exp(scores)
  __shared__ __align__(16) float qsh[DKH];
  __shared__ __align__(16) float opart[8][DKH];   // per-wave V-pass partials
  __shared__ float redmax[8];
  __shared__ float redsum[8];

  const int tid  = threadIdx.x;
  const int lane = tid & 31;
  const int wave = tid >> 5;
  const int bh   = blockIdx.x;          // b*32 + h
  const int b    = bh >> 5;
  const int h    = bh & 31;

  const int sp  = *sp_p;                // device scalar (graph-capture safe)
  const int n_s = sp + 1;               // end_pos

  const float* qrow  = q  + (size_t)b * D_MODEL + h * DKH;
  const float* knrow = kn + (size_t)b * D_MODEL + h * DKH;
  const float* vnrow = vn + (size_t)b * D_MODEL + h * DKH;
  const float* kbase = kc + (size_t)bh * SMAX * DKH;
  const float* vbase = vc + (size_t)bh * SMAX * DKH;

  if (tid < DKH) qsh[tid] = qrow[tid];
  __syncthreads();

  // ---- Pass A: raw scores -> LDS, track wave max --------------------------
  float wmax = -3.0e38f;
  #pragma unroll
  for (int i = 0; i < 8; ++i) {
    const int s = wave * 256 + i * 32 + lane;
    float val = -3.0e38f;
    if (s < n_s) {
      const float* krow = (s == sp) ? knrow : (kbase + (size_t)s * DKH);
      float dot = 0.f;
      #pragma unroll 8
      for (int d = 0; d < DKH; d += 4) {
        const float4 kv = *(const float4*)(krow + d);
        const float4 qv = *(const float4*)(qsh + d);   // LDS broadcast
        dot += kv.x * qv.x + kv.y * qv.y + kv.z * qv.z + kv.w * qv.w;
      }
      val = dot * SCALE;
    }
    sc[s] = val;
    wmax  = fmaxf(wmax, val);
  }
  for (int off = 16; off > 0; off >>= 1)
    wmax = fmaxf(wmax, __shfl_xor(wmax, off, 32));
  if (lane == 0) redmax[wave] = wmax;
  __syncthreads();
  float bmax = redmax[0];
  #pragma unroll
  for (int w = 1; w < 8; ++w) bmax = fmaxf(bmax, redmax[w]);

  // ---- exp + sum ----------------------------------------------------------
  float wsum = 0.f;
  #pragma unroll
  for (int i = 0; i < 8; ++i) {
    const int s = wave * 256 + i * 32 + lane;
    const float e = (s < n_s) ? __expf(sc[s] - bmax) : 0.f;
    sc[s] = e;
    wsum += e;
  }
  for (int off = 16; off > 0; off >>= 1)
    wsum += __shfl_xor(wsum, off, 32);
  if (lane == 0) redsum[wave] = wsum;
  __syncthreads();                      // also orders sc[] writes for Pass B
  float bsum = redsum[0];
  #pragma unroll
  for (int w = 1; w < 8; ++w) bsum += redsum[w];

  // ---- Pass B: out = sum_s attn[s] * V[s] ---------------------------------
  // Computed trip count (no mid-loop break) so the compiler can pipeline.
  const int sbase = wave * 256;
  int lim = n_s - sbase;
  lim = (lim < 0) ? 0 : (lim > 256 ? 256 : lim);

  const int d = lane * 4;
  float ax = 0.f, ay = 0.f, az = 0.f, aw = 0.f;
  #pragma unroll 8
  for (int i = 0; i < lim; ++i) {
    const int s = sbase + i;
    const float a = sc[s];               // LDS broadcast
    const float* vrow = (s == sp) ? vnrow : (vbase + (size_t)s * DKH);
    const float4 vv = *(const float4*)(vrow + d);   // full 512B row / wave
    ax += a * vv.x; ay += a * vv.y; az += a * vv.z; aw += a * vv.w;
  }
  opart[wave][d + 0] = ax;
  opart[wave][d + 1] = ay;
  opart[wave][d + 2] = az;
  opart[wave][d + 3] = aw;
  __syncthreads();

  if (tid < DKH) {
    float o = 0.f;                       // fixed-order combine: deterministic
    #pragma unroll
    for (int w = 0; w < 8; ++w) o += opart[w][tid];
    out[(size_t)b * D_MODEL + h * DKH + tid] = o / bsum;
  }
}

// ---------------------------------------------------------------------------
extern "C" void kernel_launch(void* const* d_in, const int* in_sizes, int n_in,
                              void* d_out, int out_size, void* d_ws, size_t ws_size,
                              hipStream_t stream) {
  const float* x  = (const float*)d_in[0];   // (16,1,4096)
  const float* kc = (const float*)d_in[1];   // (16,32,2048,128)
  const float* vc = (const float*)d_in[2];
  const float* Wq = (const float*)d_in[3];   // (4096,4096)
  const float* Wk = (const float*)d_in[4];
  const float* Wv = (const float*)d_in[5];
  const float* Wo = (const float*)d_in[6];
  const int*   sp = (const int*)d_in[7];     // start_pos (device scalar)
  float*       out = (float*)d_out;          // (16,1,4096)

  float* ws  = (float*)d_ws;
  float* qb  = ws;                           // [16][4096]  q,  layout [b][h*128+dk]
  float* knb = ws + 1 * NB * D_MODEL;        // [16][4096]  k_new
  float* vnb = ws + 2 * NB * D_MODEL;        // [16][4096]  v_new
  float* ab  = ws + 3 * NB * D_MODEL;        // [16][4096]  attention output

  // Fused QKV: grid (256 n-tiles, 3 weights) x 8 waves = 6144 waves
  qkv_gemm<<<dim3(D_MODEL / 16, 3), 256, 0, stream>>>(
      x, Wq, Wk, Wv, qb, knb, vnb, D_MODEL, D_MODEL);

  // Attention: one block per (b,h) = 512 blocks, 8 waves each
  attn_decode<<<NB * NH, 256, 0, stream>>>(qb, kc, vc, knb, vnb, sp, ab);

  // Output projection: 256 blocks x 8 waves = 2048 waves
  out_gemm<<<D_MODEL / 16, 256, 0, stream>>>(ab, Wo, out, D_MODEL, D_MODEL);
}